// CRF_53128745451552
// MI455X (gfx1250) — compile-verified
//
#include <hip/hip_runtime.h>
#include <hip/hip_bf16.h>

typedef __attribute__((ext_vector_type(2))) float v2f;
typedef __attribute__((ext_vector_type(8))) float v8f;

// out[b,l,i,j] = emission[b,l,j] + transition[i,j], T = 64.
//
// Each block owns ITERS consecutive (b,l) tiles; 16 waves per block; each wave
// owns one 16x16 sub-tile position (i0,j0) across all ITERS tiles.
// Per sub-tile: D = A(ones col) x B(emission row) + C(transition tile) via
// V_WMMA_F32_16X16X4_F32 (exact: 1.0*em + trans, single RNE FMA).
// A and C are loop-invariant -> loaded once per wave, reused for all ITERS tiles.
template <int ITERS>
__global__ __launch_bounds__(512) void CRF_53128745451552_kernel(
    const float* __restrict__ em,   // [B*L, 64]
    const float* __restrict__ tr,   // [64, 64]
    float* __restrict__ out,        // [B*L, 64, 64]
    int tile_base)
{
    const int tile0 = tile_base + blockIdx.x * ITERS;   // first b*L + l for this block
    const int tid   = threadIdx.x;
    const int wave  = tid >> 5;          // 0..15
    const int lane  = tid & 31;
    const int i0    = (wave >> 2) << 4;  // 0,16,32,48
    const int j0    = (wave & 3)  << 4;  // 0,16,32,48
    const int hi    = lane >> 4;         // 0: lanes 0-15, 1: lanes 16-31
    const int col   = lane & 15;

    // A matrix 16x4 f32 (2 VGPRs): VGPR0 holds K=0 (lanes 0-15) / K=2 (lanes 16-31),
    // VGPR1 holds K=1 / K=3.  We want A[m][0] = 1, rest 0.  Loop-invariant.
    v2f a;
    a.x = (hi == 0) ? 1.0f : 0.0f;
    a.y = 0.0f;

    // C matrix 16x16 f32 (8 VGPRs): VGPR r <-> row (i0 + r + 8*hi), col (j0 + col).
    // Transition is 16 KB -> L1/L2 resident; loads are coalesced (two contiguous
    // 64B runs per instruction).  Loop-invariant: loaded once, reused ITERS times.
    const float* trp = tr + (size_t)(i0 + hi * 8) * 64 + (size_t)(j0 + col);
    v8f c;
#pragma unroll
    for (int r = 0; r < 8; ++r) c[r] = trp[(size_t)r * 64];

    // Per-tile streaming pointers.
    const float* emp = em + (size_t)tile0 * 64 + (size_t)(j0 + col);
    float* op = out + (size_t)tile0 * 4096
                    + (size_t)(i0 + hi * 8) * 64 + (size_t)(j0 + col);

#pragma unroll
    for (int t = 0; t < ITERS; ++t) {
        // B matrix 4x16 f32: row K=0 = emission[tile][j0 .. j0+15].
        const float em_v = emp[(size_t)t * 64];
        v2f b;
        b.x = (hi == 0) ? em_v : 0.0f;
        b.y = 0.0f;

        // D = 1.0 * em[j] + trans[i][j]
        v8f d = __builtin_amdgcn_wmma_f32_16x16x4_f32(
            /*neg_a=*/false, a, /*neg_b=*/false, b,
            /*c_mod=*/(short)0, c, /*reuse_a=*/false, /*reuse_b=*/false);

        // Output is 256 MB (> 192 MB L2): non-temporal stores keep the resident
        // transition matrix and emission stream from being evicted.
        // Per ISA: overwriting a store's source VGPRs is not a hazard, so the
        // unrolled loop streams without s_wait_storecnt stalls.
        float* opt = op + (size_t)t * 4096;
#pragma unroll
        for (int r = 0; r < 8; ++r)
            __builtin_nontemporal_store(d[r], opt + (size_t)r * 64);
    }
}

extern "C" void kernel_launch(void* const* d_in, const int* in_sizes, int n_in,
                              void* d_out, int out_size, void* d_ws, size_t ws_size,
                              hipStream_t stream) {
    const float* em = (const float*)d_in[0];   // emission [B, L, T] f32
    const float* tr = (const float*)d_in[1];   // transition [T, T] f32
    float* out = (float*)d_out;                // [B, L, T, T] f32

    const int T  = 64;
    const int BL = in_sizes[0] / T;            // B*L = 16384

    constexpr int ITERS = 8;
    const int main_blocks = BL / ITERS;        // 2048
    const int rem         = BL - main_blocks * ITERS;

    if (main_blocks > 0) {
        CRF_53128745451552_kernel<ITERS>
            <<<dim3(main_blocks), dim3(512), 0, stream>>>(em, tr, out, 0);
    }
    if (rem > 0) {
        CRF_53128745451552_kernel<1>
            <<<dim3(rem), dim3(512), 0, stream>>>(em, tr, out, main_blocks * ITERS);
    }
}